// DownsampleBlock_83777632076468
// MI455X (gfx1250) — compile-verified
//
#include <hip/hip_runtime.h>
#include <hip/hip_bf16.h>

// ---------------------------------------------------------------------------
// Problem constants (from reference setup_inputs)
// ---------------------------------------------------------------------------
constexpr int B_  = 8;
constexpr int N_  = 8192;
constexpr int M_  = 512;    // N_CENTROIDS
constexpr int C0_ = 64;     // support channels (3 xyz + 61 features)
constexpr int H1_ = 128;
constexpr int H2_ = 256;
constexpr int D_  = 256;    // attention dim == h3

typedef __attribute__((ext_vector_type(2))) float v2f;
typedef __attribute__((ext_vector_type(8))) float v8f;

// fp32 WMMA: D(16x16) = A(16x4) * B(4x16) + C    (CDNA5 V_WMMA_F32_16X16X4_F32)
#define WMMA_F32X4(a, b, c) \
  __builtin_amdgcn_wmma_f32_16x16x4_f32(false, (a), false, (b), (short)0, (c), false, false)

// gfx1250 async global->LDS block copy (ASYNCcnt-tracked; per-lane 16B)
__device__ __forceinline__ void async_g2l_b128(uint32_t lds_byte_off, const float* gaddr) {
  asm volatile("global_load_async_to_lds_b128 %0, %1, off"
               :: "v"(lds_byte_off), "v"(gaddr)
               : "memory");
}
__device__ __forceinline__ void wait_async0() {
  asm volatile("s_wait_asynccnt 0x0" ::: "memory");
}

// ---------------------------------------------------------------------------
// 1) support = concat(xyz, point_features)  -> [B, 64, N]
// ---------------------------------------------------------------------------
__global__ void build_support_kernel(const float* __restrict__ xyz,
                                     const float* __restrict__ pf,
                                     float* __restrict__ sup) {
  size_t i = (size_t)blockIdx.x * blockDim.x + threadIdx.x;
  size_t total = (size_t)B_ * C0_ * N_;
  if (i >= total) return;
  int n = (int)(i % N_);
  int c = (int)((i / N_) % C0_);
  int b = (int)(i / ((size_t)C0_ * N_));
  float v;
  if (c < 3) v = xyz[((size_t)b * 3 + c) * N_ + n];
  else       v = pf[((size_t)b * (C0_ - 3) + (c - 3)) * N_ + n];
  sup[i] = v;
}

// [B,C,N] -> [B,N,C]  (point-major for FPS distance loops)
__global__ void transpose_kernel(const float* __restrict__ in, float* __restrict__ out) {
  size_t i = (size_t)blockIdx.x * blockDim.x + threadIdx.x;
  size_t total = (size_t)B_ * N_ * C0_;
  if (i >= total) return;
  int c = (int)(i % C0_);
  int n = (int)((i / C0_) % N_);
  int b = (int)(i / ((size_t)C0_ * N_));
  out[i] = in[((size_t)b * C0_ + c) * N_ + n];
}

// ---------------------------------------------------------------------------
// 2) Farthest point sampling. One block per batch, 256 threads.
// ---------------------------------------------------------------------------
__global__ void __launch_bounds__(256) fps_kernel(const float* __restrict__ supT, // [B,N,64]
                                                  const int* __restrict__ far0,
                                                  int* __restrict__ cidx) {
  constexpr int TPB = 256;
  constexpr int PPT = N_ / TPB; // 32 points per thread
  const int b = blockIdx.x;
  const int tid = threadIdx.x;
  const float* base = supT + (size_t)b * N_ * C0_;

  __shared__ float cen[C0_];
  __shared__ float sval[TPB];
  __shared__ int   sidx[TPB];

  float dist[PPT];
#pragma unroll
  for (int i = 0; i < PPT; i++) dist[i] = 1e10f;

  int far = far0[b];
  for (int m = 0; m < M_; m++) {
    if (tid == 0) cidx[b * M_ + m] = far;
    if (tid < C0_) cen[tid] = base[(size_t)far * C0_ + tid];
    __syncthreads();

    float best = -1.0f;
    int bidx = 0;
#pragma unroll
    for (int i = 0; i < PPT; i++) {
      int n = tid + i * TPB;
      const float* p = base + (size_t)n * C0_;
      float d = 0.0f;
#pragma unroll
      for (int c = 0; c < C0_; c += 4) {
        float4 x = *(const float4*)(p + c);
        float dx = x.x - cen[c + 0];
        float dy = x.y - cen[c + 1];
        float dz = x.z - cen[c + 2];
        float dw = x.w - cen[c + 3];
        d += dx * dx + dy * dy + dz * dz + dw * dw;
      }
      float nd = fminf(dist[i], d);
      dist[i] = nd;
      if (nd > best) { best = nd; bidx = n; }
    }
    sval[tid] = best;
    sidx[tid] = bidx;
    __syncthreads();
    for (int s = TPB / 2; s > 0; s >>= 1) {
      if (tid < s) {
        float ov = sval[tid + s]; int oi = sidx[tid + s];
        if (ov > sval[tid] || (ov == sval[tid] && oi < sidx[tid])) {
          sval[tid] = ov; sidx[tid] = oi;
        }
      }
      __syncthreads();
    }
    far = sidx[0];
    __syncthreads();
  }
}

// ---------------------------------------------------------------------------
// 3) WMMA GEMM: Y[b,o,n] = sum_c W[o,c] X[b,c,n] (+bias[o]) (+res[b,o,n])
//    fp32 A(16x4): vgpr j, lane<16 -> K=j, lane>=16 -> K=j+2, M=lane%16.
//    fp32 B(4x16): vgpr j, lane<16 -> row K=j N=lane, lane>=16 -> K=j+2.
//    fp32 C(16x16): vgpr v, lanes0-15 M=v, lanes16-31 M=v+8, N=lane%16.
// ---------------------------------------------------------------------------
__global__ void __launch_bounds__(128) wmma_gemm_kernel(
    const float* __restrict__ W, const float* __restrict__ X,
    const float* __restrict__ bias, const float* __restrict__ res,
    float* __restrict__ Y, int O, int Cin, int N, int totalWaves) {
  int wave = blockIdx.x * (blockDim.x >> 5) + (threadIdx.x >> 5);
  if (wave >= totalWaves) return;
  const int lane = threadIdx.x & 31;
  const int mrow = lane & 15;
  const int hi = lane >> 4; // which K-half this lane serves
  const int tilesN = N >> 4;
  const int tilesO = O >> 4;
  int b = wave / (tilesN * tilesO);
  int r = wave % (tilesN * tilesO);
  int o0 = (r / tilesN) << 4;
  int n0 = (r % tilesN) << 4;

  const float* Xb = X + (size_t)b * Cin * N;
  const float* wrow = W + (size_t)(o0 + mrow) * Cin + (hi << 1);
  const float* xcol = Xb + (size_t)(hi << 1) * N + n0 + mrow;

  v8f acc = {};
  for (int c0 = 0; c0 < Cin; c0 += 4) {
    if ((c0 & 31) == 0) __builtin_prefetch(xcol + (size_t)(c0 + 32) * N, 0, 1);
    v2f a, bb;
    a.x = wrow[c0];
    a.y = wrow[c0 + 1];
    bb.x = xcol[(size_t)c0 * N];
    bb.y = xcol[(size_t)c0 * N + N];
    acc = WMMA_F32X4(a, bb, acc);
  }
#pragma unroll
  for (int v = 0; v < 8; v++) {
    int o = o0 + v + (hi << 3);
    float val = acc[v];
    if (bias) val += bias[o];
    size_t oi = ((size_t)b * O + o) * N + n0 + mrow;
    if (res) val += res[oi];
    Y[oi] = val;
  }
}

// ---------------------------------------------------------------------------
// 4) BatchNorm train-mode stats + fused normalize/LeakyReLU
// ---------------------------------------------------------------------------
__global__ void __launch_bounds__(256) bn_stats_kernel(const float* __restrict__ X,
                                                       float* __restrict__ mean,
                                                       float* __restrict__ istd, int C) {
  const int c = blockIdx.x;
  const int tid = threadIdx.x;
  __shared__ float ss[256], ss2[256];
  float s = 0.f, s2 = 0.f;
  const int total = B_ * N_;
  for (int i = tid; i < total; i += 256) {
    int b = i / N_, n = i - b * N_;
    float v = X[((size_t)b * C + c) * N_ + n];
    s += v;
    s2 += v * v;
  }
  ss[tid] = s; ss2[tid] = s2;
  __syncthreads();
  for (int st = 128; st > 0; st >>= 1) {
    if (tid < st) { ss[tid] += ss[tid + st]; ss2[tid] += ss2[tid + st]; }
    __syncthreads();
  }
  if (tid == 0) {
    float m = ss[0] / (float)total;
    float var = ss2[0] / (float)total - m * m;
    mean[c] = m;
    istd[c] = rsqrtf(var + 1e-5f);
  }
}

__global__ void bn_lrelu_kernel(float* __restrict__ X, const float* __restrict__ mean,
                                const float* __restrict__ istd,
                                const float* __restrict__ gamma,
                                const float* __restrict__ beta, int C, size_t total) {
  size_t i = (size_t)blockIdx.x * blockDim.x + threadIdx.x;
  if (i >= total) return;
  int c = (int)((i / N_) % C);
  float v = (X[i] - mean[c]) * istd[c] * gamma[c] + beta[c];
  X[i] = v > 0.f ? v : 0.2f * v;
}

// ---------------------------------------------------------------------------
// 5) Gather centroids [B,256,M] from features and new_xyz [B,3,M] (to d_out).
// ---------------------------------------------------------------------------
__global__ void gather_kernel(const float* __restrict__ feat, const float* __restrict__ xyz,
                              const int* __restrict__ cidx, float* __restrict__ cent,
                              float* __restrict__ nxyz) {
  int i = blockIdx.x * blockDim.x + threadIdx.x;
  const int totalC = B_ * D_ * M_;
  if (i < totalC) {
    int b = i / (D_ * M_);
    int c = (i / M_) % D_;
    int m = i % M_;
    int idx = cidx[b * M_ + m];
    cent[i] = feat[((size_t)b * D_ + c) * N_ + idx];
  }
  const int totalX = B_ * 3 * M_;
  if (i < totalX) {
    int b = i / (3 * M_);
    int c = (i / M_) % 3;
    int m = i % M_;
    int idx = cidx[b * M_ + m];
    nxyz[i] = xyz[((size_t)b * 3 + c) * N_ + idx];
  }
}

// ---------------------------------------------------------------------------
// 6) Flash attention: one wave per 16-row centroid tile. K/V tiles are
//    double-buffered into LDS with gfx1250 async global->LDS b128 copies
//    (ASYNCcnt), overlapping the copy of tile n+1 with WMMA on tile n.
//    Row pad = 20 floats: 16B-aligned rows for b128, conflict-free PV reads.
// ---------------------------------------------------------------------------
constexpr int PAD_ = 20;

// stage one 256x16 fp32 tile (rows d, cols n0..n0+15) into sbuf via async DMA
__device__ __forceinline__ void stage_tile(const float* __restrict__ gbase, int n0,
                                           int lane, float (*sbuf)[PAD_]) {
#pragma unroll
  for (int i = 0; i < 32; i++) {
    int chunk = i * 32 + lane;   // 0..1023, 16B each
    int d = chunk >> 2;
    int sub = chunk & 3;
    const float* g = gbase + (size_t)d * N_ + n0 + sub * 4;
    uint32_t lds = (uint32_t)(size_t)(&sbuf[d][sub * 4]);
    async_g2l_b128(lds, g);
  }
}

__global__ void __launch_bounds__(32) flash_attn_kernel(
    const float* __restrict__ Q, const float* __restrict__ K,
    const float* __restrict__ V, float* __restrict__ Aout) {
  const int b = blockIdx.y;
  const int m0 = blockIdx.x << 4;
  const int lane = threadIdx.x;
  const int mrow = lane & 15;
  const int hi = lane >> 4;
  const int h2 = hi << 1;

  __shared__ __attribute__((aligned(16))) float sQ[D_][16];      // Q^T tile: sQ[d][m]
  __shared__ __attribute__((aligned(16))) float sK[2][D_][PAD_]; // K tiles (dbl-buffered)
  __shared__ __attribute__((aligned(16))) float sV[2][D_][PAD_]; // V tiles (dbl-buffered)
  __shared__ float sP[16][17];                                   // probs tile
  __shared__ float sMax[16], sSum[16], sFac[16];

  const float* Qb = Q + (size_t)b * D_ * M_;
  const float* Kb = K + (size_t)b * D_ * N_;
  const float* Vb = V + (size_t)b * D_ * N_;

  // kick off async staging of tile 0 before doing anything else
  stage_tile(Kb, 0, lane, sK[0]);
  stage_tile(Vb, 0, lane, sV[0]);

  for (int i = lane; i < D_ * 16; i += 32) {
    int d = i >> 4, m = i & 15;
    sQ[d][m] = Qb[(size_t)d * M_ + m0 + m];
  }
  if (lane < 16) { sMax[lane] = -3.0e38f; sSum[lane] = 0.f; }

  v8f acc[16];
#pragma unroll
  for (int t = 0; t < 16; t++) acc[t] = (v8f){};
  __syncthreads();

  const float scale = 0.0625f; // 1/sqrt(256)
  int cur = 0;

  for (int n0 = 0; n0 < N_; n0 += 16) {
    wait_async0();               // tile n0 resident in sK[cur]/sV[cur]
    if (n0 + 16 < N_) {          // overlap: stream tile n0+16 into other buffer
      stage_tile(Kb, n0 + 16, lane, sK[cur ^ 1]);
      stage_tile(Vb, n0 + 16, lane, sV[cur ^ 1]);
    }

    // ---- S = (Q^T K) tile, 64 wmma over K=256 (A from sQ, B from sK)
    v8f s = {};
#pragma unroll 8
    for (int c0 = 0; c0 < D_; c0 += 4) {
      v2f a, bb;
      a.x = sQ[c0 + h2][mrow];
      a.y = sQ[c0 + h2 + 1][mrow];
      bb.x = sK[cur][c0 + h2][mrow];
      bb.y = sK[cur][c0 + h2 + 1][mrow];
      s = WMMA_F32X4(a, bb, s);
    }
#pragma unroll
    for (int v = 0; v < 8; v++) sP[v + (hi << 3)][mrow] = s[v] * scale;
    __syncthreads();

    // ---- online softmax bookkeeping (lanes 0-15 each own a row)
    if (lane < 16) {
      float oldm = sMax[lane];
      float rm = oldm;
#pragma unroll
      for (int j = 0; j < 16; j++) rm = fmaxf(rm, sP[lane][j]);
      float fac = __expf(oldm - rm);
      float rs = 0.f;
#pragma unroll
      for (int j = 0; j < 16; j++) {
        float p = __expf(sP[lane][j] - rm);
        sP[lane][j] = p;
        rs += p;
      }
      sSum[lane] = sSum[lane] * fac + rs;
      sMax[lane] = rm;
      sFac[lane] = fac;
    }
    __syncthreads();

    // ---- rescale accumulators by per-row correction factor
#pragma unroll
    for (int v = 0; v < 8; v++) {
      float f = sFac[v + (hi << 3)];
#pragma unroll
      for (int t = 0; t < 16; t++) acc[t][v] *= f;
    }

    // ---- acc[m, d] += P(16x16) * V^T(16x16) per 16-wide d tile (64 wmma)
    for (int dt = 0; dt < 16; dt++) {
#pragma unroll
      for (int kk = 0; kk < 16; kk += 4) {
        v2f a, bb;
        a.x = sP[mrow][kk + h2];
        a.y = sP[mrow][kk + h2 + 1];
        bb.x = sV[cur][dt * 16 + mrow][kk + h2];
        bb.y = sV[cur][dt * 16 + mrow][kk + h2 + 1];
        acc[dt] = WMMA_F32X4(a, bb, acc[dt]);
      }
    }
    __syncthreads();
    cur ^= 1;
  }

  float inv[8];
#pragma unroll
  for (int v = 0; v < 8; v++) inv[v] = 1.0f / sSum[v + (hi << 3)];
#pragma unroll
  for (int dt = 0; dt < 16; dt++) {
#pragma unroll
    for (int v = 0; v < 8; v++) {
      int d = dt * 16 + mrow;
      Aout[((size_t)b * D_ + d) * M_ + m0 + v + (hi << 3)] = acc[dt][v] * inv[v];
    }
  }
}

// ---------------------------------------------------------------------------
// Host launcher
// ---------------------------------------------------------------------------
static inline void launch_gemm(const float* W, const float* X, const float* bias,
                               const float* res, float* Y, int O, int Cin, int N,
                               hipStream_t stream) {
  int totalWaves = B_ * (O / 16) * (N / 16);
  int blocks = (totalWaves + 3) / 4;
  wmma_gemm_kernel<<<blocks, 128, 0, stream>>>(W, X, bias, res, Y, O, Cin, N, totalWaves);
}

extern "C" void kernel_launch(void* const* d_in, const int* in_sizes, int n_in,
                              void* d_out, int out_size, void* d_ws, size_t ws_size,
                              hipStream_t stream) {
  const float* xyz = (const float*)d_in[0];
  const float* pf  = (const float*)d_in[1];
  const float* w1  = (const float*)d_in[2];
  const float* b1  = (const float*)d_in[3];
  const float* g1  = (const float*)d_in[4];
  const float* be1 = (const float*)d_in[5];
  const float* w2  = (const float*)d_in[6];
  const float* b2  = (const float*)d_in[7];
  const float* g2  = (const float*)d_in[8];
  const float* be2 = (const float*)d_in[9];
  const float* w3  = (const float*)d_in[10];
  const float* b3  = (const float*)d_in[11];
  const float* wq  = (const float*)d_in[12];
  const float* wk  = (const float*)d_in[13];
  const float* wv  = (const float*)d_in[14];
  const float* wo  = (const float*)d_in[15];
  const int*  far0 = (const int*)d_in[16];
  float* out = (float*)d_out;
  float* ws = (float*)d_ws;

  // workspace layout (floats)
  const size_t SUP  = 0;                                  // [B,64,N]
  const size_t SUPT = SUP  + (size_t)B_ * C0_ * N_;       // [B,N,64]
  const size_t CIDX = SUPT + (size_t)B_ * C0_ * N_;       // [B,M] int
  const size_t MEAN = CIDX + (size_t)B_ * M_;             // [<=256]
  const size_t ISTD = MEAN + 256;                         // [<=256]
  const size_t CENT = ISTD + 256;                         // [B,256,M]
  const size_t QB   = CENT + (size_t)B_ * D_ * M_;        // [B,256,M]
  const size_t ATT  = QB   + (size_t)B_ * D_ * M_;        // [B,256,M]
  const size_t BUFA = ATT  + (size_t)B_ * D_ * M_;        // [B,256,N]
  const size_t BUFB = BUFA + (size_t)B_ * H2_ * N_;       // [B,256,N]
  const size_t BUFC = BUFB + (size_t)B_ * H2_ * N_;       // [B,256,N]
  int* cidx = (int*)(ws + CIDX);

  // 1) support + transpose + FPS
  {
    size_t total = (size_t)B_ * C0_ * N_;
    int blocks = (int)((total + 255) / 256);
    build_support_kernel<<<blocks, 256, 0, stream>>>(xyz, pf, ws + SUP);
    transpose_kernel<<<blocks, 256, 0, stream>>>(ws + SUP, ws + SUPT);
  }
  fps_kernel<<<B_, 256, 0, stream>>>(ws + SUPT, far0, cidx);

  // 2) MLP: conv1 -> BN -> LReLU -> conv2 -> BN -> LReLU -> conv3
  launch_gemm(w1, ws + SUP, b1, nullptr, ws + BUFA, H1_, C0_, N_, stream);
  bn_stats_kernel<<<H1_, 256, 0, stream>>>(ws + BUFA, ws + MEAN, ws + ISTD, H1_);
  {
    size_t total = (size_t)B_ * H1_ * N_;
    bn_lrelu_kernel<<<(int)((total + 255) / 256), 256, 0, stream>>>(
        ws + BUFA, ws + MEAN, ws + ISTD, g1, be1, H1_, total);
  }
  launch_gemm(w2, ws + BUFA, b2, nullptr, ws + BUFB, H2_, H1_, N_, stream);
  bn_stats_kernel<<<H2_, 256, 0, stream>>>(ws + BUFB, ws + MEAN, ws + ISTD, H2_);
  {
    size_t total = (size_t)B_ * H2_ * N_;
    bn_lrelu_kernel<<<(int)((total + 255) / 256), 256, 0, stream>>>(
        ws + BUFB, ws + MEAN, ws + ISTD, g2, be2, H2_, total);
  }
  launch_gemm(w3, ws + BUFB, b3, nullptr, ws + BUFA, H2_, H2_, N_, stream); // features

  // 3) gather centroids + new_xyz (new_xyz straight into d_out)
  gather_kernel<<<(B_ * D_ * M_ + 255) / 256, 256, 0, stream>>>(
      ws + BUFA, xyz, cidx, ws + CENT, out);

  // 4) q/k/v projections
  launch_gemm(wq, ws + CENT, nullptr, nullptr, ws + QB,   D_, D_, M_, stream);
  launch_gemm(wk, ws + BUFA, nullptr, nullptr, ws + BUFB, D_, D_, N_, stream);
  launch_gemm(wv, ws + BUFA, nullptr, nullptr, ws + BUFC, D_, D_, N_, stream);

  // 5) flash attention (async-DMA double-buffered K/V tiles)
  flash_attn_kernel<<<dim3(M_ / 16, B_), 32, 0, stream>>>(
      ws + QB, ws + BUFB, ws + BUFC, ws + ATT);

  // 6) out2 = wo * att + centroids
  launch_gemm(wo, ws + ATT, nullptr, ws + CENT, out + (size_t)B_ * 3 * M_,
              D_, D_, M_, stream);
}